// SCANCircuitV4B_27144193310728
// MI455X (gfx1250) — compile-verified
//
#include <hip/hip_runtime.h>

typedef __attribute__((ext_vector_type(2))) float v2f;
typedef __attribute__((ext_vector_type(4))) float v4f;
typedef __attribute__((ext_vector_type(8))) float v8f;

#define MO 64
#define NA 8

__device__ __forceinline__ int clampi(int x, int lo, int hi) {
    return x < lo ? lo : (x > hi ? hi : x);
}

// One wave (32 threads) per batch element.
//
// Only the root node's buffer reaches the output, and every nonzero row of it
// is exactly one embedding vector. The final row pattern is always
//   rows [0,c1) -> E[k1], rows [c1,c1+c2) -> E[k2], rest zero,  rc = c1+c2 <= 6.
// So logits rows 0..15 are one 16x16 WMMA tile of R(16x32) @ E^T(32x16),
// rows 16..63 are the constant stop-bias pattern.
//
// No operand masking is needed for the WMMA:
//  - B columns 8..15 mirror columns 0..7 (never stored, acol<8 guard)
//  - A rows >= rc hold a real (garbage) embedding row; zeroed at the store.
__global__ void scan_logits_kernel(const int* __restrict__ cats,
                                   const int* __restrict__ subs,
                                   const unsigned char* __restrict__ maskb,
                                   const int* __restrict__ cl,
                                   const int* __restrict__ cr,
                                   const float* __restrict__ E,
                                   float* __restrict__ out)
{
    const int b    = blockIdx.x;
    const int base = b * 64;
    const int lane = threadIdx.x;   // 0..31, wave32

    // ---- unconditional field loads (uniform values across the wave) ----
    const int c0 = cats[base];
    const int s0 = subs[base];
    const int m0 = maskb[base];
    const int l  = clampi(cl[base], 0, 63);
    const int r  = clampi(cr[base], 0, 63);

    const int cL = cats[base + l], sL = subs[base + l], mL = maskb[base + l];
    const int cR = cats[base + r], sR = subs[base + r], mR = maskb[base + r];
    const int gL = clampi(cl[base + l], 0, 63);
    const int gR = clampi(cl[base + r], 0, 63);

    const int cGL = cats[base + gL], sGL = subs[base + gL], mGL = maskb[base + gL];
    const int cGR = cats[base + gR], sGR = subs[base + gR], mGR = maskb[base + gR];

    // ---- branch-free tree decode ----
    const bool primL  = (cL == 0) & (mL != 0);
    const bool modL   = (cL == 1) & (mL != 0);
    const bool primGL = (cGL == 0) & (mGL != 0);
    const bool primR  = (cR == 0) & (mR != 0);
    const bool modR   = (cR == 1) & (mR != 0);
    const bool primGR = (cGR == 0) & (mGR != 0);

    // post-mod(left child): count and replicated action index
    const int pmL_c = primL ? 1 : ((modL && primGL) ? clampi(sL + 2, 1, 3) : 0);
    const int pmL_k = primL ? clampi(sL + 1, 0, 7) : clampi(sGL + 1, 0, 7);
    // post-mod(right child)
    const int pmR_c = primR ? 1 : ((modR && primGR) ? clampi(sR + 2, 1, 3) : 0);
    const int pmR_k = primR ? clampi(sR + 1, 0, 7) : clampi(sGR + 1, 0, 7);

    const bool rootPrim = (c0 == 0) & (m0 != 0);
    const bool rootMod  = (c0 == 1) & (m0 != 0);
    const bool rootComb = (c0 == 2) & (m0 != 0);
    const bool after    = (s0 == 1);

    const int f_c = after ? pmR_c : pmL_c;
    const int f_k = after ? pmR_k : pmL_k;
    const int s_c = after ? pmL_c : pmR_c;
    const int s_k = after ? pmL_k : pmR_k;

    const int c1 = rootPrim ? 1
                 : (rootMod  ? (primL ? clampi(s0 + 2, 1, 3) : 0)
                 : (rootComb ? f_c : 0));
    const int k1 = rootPrim ? clampi(s0 + 1, 0, 7)
                 : (rootMod  ? clampi(sL + 1, 0, 7) : f_k);
    const int c2 = rootComb ? s_c : 0;
    const int k2 = s_k;

    int rc = c1 + c2; if (rc > 64) rc = 64;   // root count (<= 6 in practice)

    // ---- WMMA: C(16x16) = A(16x4) x B(4x16), accumulated over 8 K-steps ----
    // 32-bit A layout: lane 0-15 -> M=lane, K={0,1}; lane 16-31 -> M=lane-16, K={2,3}.
    // B mirrors with N in place of M.
    const int mrow  = lane & 15;
    const int khalf = (lane >> 4) << 1;            // 0 or 2
    const int kidx  = (mrow < c1) ? k1 : k2;       // rows >= rc masked at store
    const int nidx  = lane & 7;                    // cols 8..15 mirror, never stored

    const float* __restrict__ Arow = E + kidx * 32 + khalf;  // 8B aligned (khalf even)
    const float* __restrict__ Brow = E + nidx * 32 + khalf;

    v8f acc = {};
#pragma unroll
    for (int kk = 0; kk < 8; ++kk) {
        const v2f a  = *(const v2f*)(Arow + kk * 4);
        const v2f bb = *(const v2f*)(Brow + kk * 4);
        acc = __builtin_amdgcn_wmma_f32_16x16x4_f32(
                  /*neg_a=*/false, a, /*neg_b=*/false, bb,
                  /*c_mod=*/(short)0, acc, /*reuse_a=*/false, /*reuse_b=*/false);
    }

    // ---- store ----
    // C layout: VGPR j, lanes 0-15 -> (M=j, N=lane); lanes 16-31 -> (M=8+j, N=lane-16)
    float* __restrict__ outb = out + b * (MO * NA);
    const int acol  = lane & 15;                   // action column
    const int pbase = (lane >> 4) << 3;            // 0 or 8
    if (acol < 8) {
#pragma unroll
        for (int j = 0; j < 8; ++j) {
            const int pos = pbase + j;
            float v = (pos < rc) ? acc[j] : 0.0f;          // zero garbage rows
            if (acol == 7 && pos >= rc) v += 8.0f;         // stop bias
            __builtin_nontemporal_store(v, outb + pos * NA + acol);
        }
    }

    // positions 16..63: zero rows + stop bias, 3 x b128 per lane, coalesced
    const v4f zv = {0.0f, 0.0f, 0.0f, (lane & 1) ? 8.0f : 0.0f};
#pragma unroll
    for (int i = 0; i < 3; ++i) {
        __builtin_nontemporal_store(zv, (v4f*)(outb + 128 + i * 128 + lane * 4));
    }
}

extern "C" void kernel_launch(void* const* d_in, const int* in_sizes, int n_in,
                              void* d_out, int out_size, void* d_ws, size_t ws_size,
                              hipStream_t stream)
{
    const int*           cats  = (const int*)d_in[0];
    const int*           subs  = (const int*)d_in[1];
    const unsigned char* maskb = (const unsigned char*)d_in[2]; // jnp bool = 1 byte
    const int*           cl    = (const int*)d_in[3];
    const int*           cr    = (const int*)d_in[4];
    const float*         E     = (const float*)d_in[5];
    float*               outp  = (float*)d_out;

    const int B = in_sizes[0] / 64;                // 512 batches
    scan_logits_kernel<<<B, 32, 0, stream>>>(cats, subs, maskb, cl, cr, E, outp);
}